// AutoLink_l3Rs_33998961116076
// MI455X (gfx1250) — compile-verified
//
#include <hip/hip_runtime.h>
#include <hip/hip_bf16.h>

typedef __attribute__((ext_vector_type(2))) float v2f;
typedef __attribute__((ext_vector_type(8))) float v8f;

#define DCH 128
#define LDST 130   // padded LDS row stride (even -> 8B-aligned b64 reads, fewer bank conflicts)

// ---------------- utility kernels ----------------

__global__ void fill0_f4(float4* __restrict__ p, int n4) {
    int t = blockIdx.x * blockDim.x + threadIdx.x;
    if (t < n4) p[t] = float4{0.f, 0.f, 0.f, 0.f};
}

__global__ void copy_f4(float4* __restrict__ dst, const float4* __restrict__ src, int n4) {
    int t = blockIdx.x * blockDim.x + threadIdx.x;
    if (t < n4) dst[t] = src[t];
}

__global__ void deg_accum(const int* __restrict__ edge_dst, float* __restrict__ deg, int E) {
    int t = blockIdx.x * blockDim.x + threadIdx.x;
    if (t < E) atomicAdd(&deg[edge_dst[t]], 1.0f);
}

__global__ void deg_invert(float* __restrict__ deg, int N) {
    int t = blockIdx.x * blockDim.x + threadIdx.x;
    if (t < N) {
        float d = deg[t];
        deg[t] = (d > 0.f) ? (1.0f / d) : 0.0f;
    }
}

// one wave per edge; lane handles 4 contiguous channels (float4 gather + 4 f32 atomics)
__global__ void sage_scatter(const float* __restrict__ h,
                             const int* __restrict__ edge_src,
                             const int* __restrict__ edge_dst,
                             float* __restrict__ agg, int E) {
    int e = blockIdx.x * (blockDim.x >> 5) + (threadIdx.x >> 5);
    int lane = threadIdx.x & 31;
    if (e >= E) return;
    int s = edge_src[e];
    int d = edge_dst[e];
    float4 v = *((const float4*)(h + (size_t)s * DCH) + lane);
    float* p = agg + (size_t)d * DCH + lane * 4;
    atomicAdd(p + 0, v.x);
    atomicAdd(p + 1, v.y);
    atomicAdd(p + 2, v.z);
    atomicAdd(p + 3, v.w);
}

// ---------------- fused SAGE layer GEMM: out = (agg*inv_deg)@Wl + bl + h@Wr ----------------
// 4 waves/block, wave computes 16 rows x 128 cols with V_WMMA_F32_16X16X4_F32.
// Weights staged transposed in LDS so B fragments are single ds_load_b64 per lane.

__global__ __launch_bounds__(128) void sage_gemm_wmma(
    const float* __restrict__ agg, const float* __restrict__ invdeg,
    float* __restrict__ h,
    const float* __restrict__ Wl, const float* __restrict__ bl,
    const float* __restrict__ Wr,
    float* __restrict__ out, int M, int layer) {

    __shared__ float lds[2 * DCH * LDST];
    float* WlT = lds;
    float* WrT = lds + DCH * LDST;

    // stage transposed weights: WxT[n][k] = Wx[k][n]
    for (int idx = threadIdx.x; idx < DCH * DCH; idx += 128) {
        int k = idx >> 7;
        int n = idx & (DCH - 1);
        WlT[n * LDST + k] = Wl[idx];
        WrT[n * LDST + k] = Wr[idx];
    }
    __syncthreads();

    const int wave = threadIdx.x >> 5;
    const int lane = threadIdx.x & 31;
    const int hi   = lane >> 4;      // 0: K-pair {0,1}/rows M+0..7 ; 1: K-pair {2,3}/rows M+8..15
    const int l16  = lane & 15;

    const int r0 = (blockIdx.x * 4 + wave) * 16;
    int arow = r0 + l16;
    if (arow >= M) arow = M - 1;     // clamp loads; WMMA requires EXEC all-1s

    const float s = invdeg[arow];

    v8f acc[8] = {};                 // 8 N-tiles of 16x16 f32 accumulators

    // pass 1: A = agg * inv_deg, B = WlT
    const float* A1 = agg + (size_t)arow * DCH;
    for (int kb = 0; kb < DCH; kb += 4) {
        const int kk = kb + 2 * hi;
        v2f a = *(const v2f*)(A1 + kk);
        a *= s;
#pragma unroll
        for (int nt = 0; nt < 8; ++nt) {
            v2f b = *(const v2f*)(WlT + (nt * 16 + l16) * LDST + kk);
            acc[nt] = __builtin_amdgcn_wmma_f32_16x16x4_f32(
                false, a, false, b, (short)0, acc[nt], false, false);
        }
    }
    // pass 2: A = h, B = WrT
    const float* A2 = h + (size_t)arow * DCH;
    for (int kb = 0; kb < DCH; kb += 4) {
        const int kk = kb + 2 * hi;
        v2f a = *(const v2f*)(A2 + kk);
#pragma unroll
        for (int nt = 0; nt < 8; ++nt) {
            v2f b = *(const v2f*)(WrT + (nt * 16 + l16) * LDST + kk);
            acc[nt] = __builtin_amdgcn_wmma_f32_16x16x4_f32(
                false, a, false, b, (short)0, acc[nt], false, false);
        }
    }

    // epilogue: bias, store raw to out[:, layer, :], relu back into h (in-place, own rows only)
#pragma unroll
    for (int nt = 0; nt < 8; ++nt) {
        const int col = nt * 16 + l16;
        const float bv = bl[col];
#pragma unroll
        for (int r = 0; r < 8; ++r) {
            int row = r0 + r + 8 * hi;
            if (row < M) {
                float v = acc[nt][r] + bv;
                out[(size_t)row * (3 * DCH) + (size_t)layer * DCH + col] = v;
                h[(size_t)row * DCH + col] = v > 0.f ? v : 0.f;
            }
        }
    }
}

// ---------------- launch ----------------

extern "C" void kernel_launch(void* const* d_in, const int* in_sizes, int n_in,
                              void* d_out, int out_size, void* d_ws, size_t ws_size,
                              hipStream_t stream) {
    const float* x   = (const float*)d_in[0];
    const float* Wl  = (const float*)d_in[1];
    const float* bl  = (const float*)d_in[2];
    const float* Wr  = (const float*)d_in[3];
    const int* esrc  = (const int*)d_in[4];
    const int* edst  = (const int*)d_in[5];
    float* out = (float*)d_out;

    const int N = in_sizes[0] / DCH;   // 50000
    const int E = in_sizes[4];         // 800000
    const int NLAYERS = 3;

    float* h   = (float*)d_ws;                       // N*128 f32
    float* agg = h + (size_t)N * DCH;                // N*128 f32
    float* inv = agg + (size_t)N * DCH;              // N f32

    const int threads = 256;
    const int nd4 = (N * DCH) / 4;

    // degrees -> inverse degrees (in place)
    fill0_f4<<<(N / 4 + threads - 1) / threads, threads, 0, stream>>>((float4*)inv, N / 4);
    deg_accum<<<(E + threads - 1) / threads, threads, 0, stream>>>(edst, inv, E);
    deg_invert<<<(N + threads - 1) / threads, threads, 0, stream>>>(inv, N);

    // h = x
    copy_f4<<<(nd4 + threads - 1) / threads, threads, 0, stream>>>(
        (float4*)h, (const float4*)x, nd4);

    const int edgeBlocks = (E + (threads / 32) - 1) / (threads / 32);
    const int gemmBlocks = (N + 63) / 64;

    for (int l = 0; l < NLAYERS; ++l) {
        fill0_f4<<<(nd4 + threads - 1) / threads, threads, 0, stream>>>((float4*)agg, nd4);
        sage_scatter<<<edgeBlocks, threads, 0, stream>>>(h, esrc, edst, agg, E);
        sage_gemm_wmma<<<gemmBlocks, 128, 0, stream>>>(
            agg, inv, h,
            Wl + (size_t)l * DCH * DCH, bl + (size_t)l * DCH,
            Wr + (size_t)l * DCH * DCH,
            out, N, l);
    }
}